// CELCLoss_80247168958715
// MI455X (gfx1250) — compile-verified
//
#include <hip/hip_runtime.h>
#include <hip/hip_bf16.h>

#define IGNORE_IDX (-100)
#define NBLOCKS 3072
#define TPB 256
#define WAVES_PER_BLOCK (TPB / 32)
#define MINIT (-3.0e38f)

// ---------- fast hw transcendentals (v_exp_f32 / v_log_f32 are base-2) ----
__device__ __forceinline__ float fast_exp2(float x) {
#if __has_builtin(__builtin_amdgcn_exp2f)
  return __builtin_amdgcn_exp2f(x);
#else
  return exp2f(x);
#endif
}
__device__ __forceinline__ float fast_log2(float x) {
#if __has_builtin(__builtin_amdgcn_logf)
  return __builtin_amdgcn_logf(x);
#else
  return log2f(x);
#endif
}

// online logsumexp pair combine: (m,s) <- combine((m,s),(m2,s2)), base-2 domain
__device__ __forceinline__ void lse_combine(float &m, float &s, float m2, float s2) {
  float mm = fmaxf(m, m2);
  s = s * fast_exp2(m - mm) + s2 * fast_exp2(m2 - mm);
  m = mm;
}

// ---------- CDNA5 async global->LDS staging (gfx1250, ASYNCcnt) -----------
#if defined(__AMDGCN__) && __has_builtin(__builtin_amdgcn_global_load_async_to_lds_b128)
#define USE_ASYNC_LDS 1
#endif

#if defined(USE_ASYNC_LDS)
// builtin prototype (confirmed by clang): (v4i AS1*, v4i AS3*, imm offset, imm cpol)
typedef int v4i_cd5 __attribute__((__vector_size__(4 * sizeof(int))));
typedef __attribute__((address_space(1))) v4i_cd5 GV4;
typedef __attribute__((address_space(3))) v4i_cd5 LV4;
__device__ __forceinline__ void async_cp16(const float4 *g, float4 *l) {
  // per-lane 16B DMA: copies 16 bytes from global address g to LDS address l; tracked by ASYNCcnt
  __builtin_amdgcn_global_load_async_to_lds_b128((GV4 *)g, (LV4 *)l, 0, 0);
}
__device__ __forceinline__ void wait_async_le4() {
#if __has_builtin(__builtin_amdgcn_s_wait_asynccnt)
  __builtin_amdgcn_s_wait_asynccnt(4);
#else
  asm volatile("s_wait_asynccnt 4" ::: "memory");
#endif
}
__device__ __forceinline__ void wait_async_le2() {
#if __has_builtin(__builtin_amdgcn_s_wait_asynccnt)
  __builtin_amdgcn_s_wait_asynccnt(2);
#else
  asm volatile("s_wait_asynccnt 2" ::: "memory");
#endif
}
__device__ __forceinline__ void wait_async_le0() {
#if __has_builtin(__builtin_amdgcn_s_wait_asynccnt)
  __builtin_amdgcn_s_wait_asynccnt(0);
#else
  asm volatile("s_wait_asynccnt 0" ::: "memory");
#endif
}
#endif

// ===========================================================================
// Kernel 1: per-block online logsumexp partials of w = (1+lam)*xn - lam*xb
// computed in base-2 domain: u = c1*xn + c2*xb,  c1=(1+lam)*log2e, c2=-lam*log2e
// ===========================================================================
__global__ __launch_bounds__(TPB) void celc_lse_partial(
    const float *__restrict__ xn, const float *__restrict__ xb,
    const float *__restrict__ lamp, float *__restrict__ part_m,
    float *__restrict__ part_s, long long n4, long long ntotal) {
  const float lam = lamp[0];
  const float L2E = 1.4426950408889634f;
  const float c1 = (1.0f + lam) * L2E;
  const float c2 = -lam * L2E;

  const float4 *gn = (const float4 *)xn;
  const float4 *gb = (const float4 *)xb;

  const long long tid0 = (long long)blockIdx.x * TPB + threadIdx.x;
  const long long stride = (long long)gridDim.x * TPB;

  float m = MINIT, s = 0.0f;

#if defined(USE_ASYNC_LDS)
  // distance-2 software pipeline, triple-buffered in LDS (24 KB / block)
  __shared__ float4 sN[3][TPB];
  __shared__ float4 sB[3][TPB];
  if (tid0 < n4) {
    async_cp16(&gn[tid0], &sN[0][threadIdx.x]);
    async_cp16(&gb[tid0], &sB[0][threadIdx.x]);
  }
  if (tid0 + stride < n4) {
    async_cp16(&gn[tid0 + stride], &sN[1][threadIdx.x]);
    async_cp16(&gb[tid0 + stride], &sB[1][threadIdx.x]);
  }
  int cur = 0;
  for (long long i = tid0; i < n4; i += stride) {
    const long long nx1 = i + stride;
    const long long nx2 = i + 2 * stride;
    int nxbuf = cur + 2;
    if (nxbuf >= 3) nxbuf -= 3;
    if (nx2 < n4) {  // stage tile i+2, then wait for tile i (async loads in-order)
      async_cp16(&gn[nx2], &sN[nxbuf][threadIdx.x]);
      async_cp16(&gb[nx2], &sB[nxbuf][threadIdx.x]);
      wait_async_le4();
    } else if (nx1 < n4) {
      wait_async_le2();
    } else {
      wait_async_le0();
    }
    const float4 vn = sN[cur][threadIdx.x];  // own lane's staged 16B
    const float4 vb = sB[cur][threadIdx.x];
    cur += 1;
    if (cur >= 3) cur -= 3;
    float u0 = fmaf(c1, vn.x, c2 * vb.x);
    float u1 = fmaf(c1, vn.y, c2 * vb.y);
    float u2 = fmaf(c1, vn.z, c2 * vb.z);
    float u3 = fmaf(c1, vn.w, c2 * vb.w);
    float mu = fmaxf(fmaxf(u0, u1), fmaxf(u2, u3));
    float mn = fmaxf(m, mu);
    s = s * fast_exp2(m - mn) + (fast_exp2(u0 - mn) + fast_exp2(u1 - mn)) +
        (fast_exp2(u2 - mn) + fast_exp2(u3 - mn));
    m = mn;
  }
#else
  for (long long i = tid0; i < n4; i += stride) {
    const float4 vn = gn[i];
    const float4 vb = gb[i];
    float u0 = fmaf(c1, vn.x, c2 * vb.x);
    float u1 = fmaf(c1, vn.y, c2 * vb.y);
    float u2 = fmaf(c1, vn.z, c2 * vb.z);
    float u3 = fmaf(c1, vn.w, c2 * vb.w);
    float mu = fmaxf(fmaxf(u0, u1), fmaxf(u2, u3));
    float mn = fmaxf(m, mu);
    s = s * fast_exp2(m - mn) + (fast_exp2(u0 - mn) + fast_exp2(u1 - mn)) +
        (fast_exp2(u2 - mn) + fast_exp2(u3 - mn));
    m = mn;
  }
#endif

  // scalar tail (if total not divisible by 4) — single thread, tiny
  if (blockIdx.x == 0 && threadIdx.x == 0) {
    for (long long e = n4 * 4; e < ntotal; ++e) {
      float u = fmaf(c1, xn[e], c2 * xb[e]);
      float mn = fmaxf(m, u);
      s = s * fast_exp2(m - mn) + fast_exp2(u - mn);
      m = mn;
    }
  }

  // wave32 reduction of (m,s)
  for (int off = 16; off > 0; off >>= 1) {
    float m2 = __shfl_xor(m, off, 32);
    float s2 = __shfl_xor(s, off, 32);
    lse_combine(m, s, m2, s2);
  }
  __shared__ float wm[WAVES_PER_BLOCK], wsv[WAVES_PER_BLOCK];
  const int lane = threadIdx.x & 31;
  const int wid = threadIdx.x >> 5;
  if (lane == 0) { wm[wid] = m; wsv[wid] = s; }
  __syncthreads();
  if (threadIdx.x == 0) {
    float M = wm[0], S = wsv[0];
    for (int w = 1; w < WAVES_PER_BLOCK; ++w) lse_combine(M, S, wm[w], wsv[w]);
    part_m[blockIdx.x] = M;
    part_s[blockIdx.x] = S;
  }
}

// ===========================================================================
// Kernel 2: combine partials -> scalar natural-log LSE
// ===========================================================================
__global__ __launch_bounds__(TPB) void celc_lse_final(
    const float *__restrict__ part_m, const float *__restrict__ part_s,
    float *__restrict__ lse_out, int nparts) {
  float m = MINIT, s = 0.0f;
  for (int j = threadIdx.x; j < nparts; j += TPB)
    lse_combine(m, s, part_m[j], part_s[j]);
  for (int off = 16; off > 0; off >>= 1) {
    float m2 = __shfl_xor(m, off, 32);
    float s2 = __shfl_xor(s, off, 32);
    lse_combine(m, s, m2, s2);
  }
  __shared__ float wm[WAVES_PER_BLOCK], wsv[WAVES_PER_BLOCK];
  const int lane = threadIdx.x & 31;
  const int wid = threadIdx.x >> 5;
  if (lane == 0) { wm[wid] = m; wsv[wid] = s; }
  __syncthreads();
  if (threadIdx.x == 0) {
    float M = wm[0], S = wsv[0];
    for (int w = 1; w < WAVES_PER_BLOCK; ++w) lse_combine(M, S, wm[w], wsv[w]);
    const float LN2 = 0.6931471805599453f;
    lse_out[0] = (M + fast_log2(S)) * LN2;  // ln(sum exp(w)) in natural log
  }
}

// ===========================================================================
// Kernel 3: gather picked logits, mean NLL over valid targets
// ===========================================================================
__global__ __launch_bounds__(TPB) void celc_gather_loss(
    const float *__restrict__ xn, const float *__restrict__ xb,
    const int *__restrict__ tgt, const float *__restrict__ lamp,
    const float *__restrict__ lse_p, float *__restrict__ out, int N,
    long long C) {
  const float lam = lamp[0];
  const float lse = lse_p[0];
  float sum = 0.0f, cnt = 0.0f;
  for (int i = threadIdx.x; i < N; i += TPB) {
    const int t = tgt[i];
    if (t != IGNORE_IDX) {
      const long long off = (long long)i * C + (long long)t;
      const float a = xn[off];
      const float b = xb[off];
      const float w = fmaf(lam, a - b, a);  // (1+lam)*a - lam*b
      sum += (w - lse);
      cnt += 1.0f;
    }
  }
  for (int off = 16; off > 0; off >>= 1) {
    sum += __shfl_xor(sum, off, 32);
    cnt += __shfl_xor(cnt, off, 32);
  }
  __shared__ float wsum[WAVES_PER_BLOCK], wcnt[WAVES_PER_BLOCK];
  const int lane = threadIdx.x & 31;
  const int wid = threadIdx.x >> 5;
  if (lane == 0) { wsum[wid] = sum; wcnt[wid] = cnt; }
  __syncthreads();
  if (threadIdx.x == 0) {
    float S = 0.0f, Cv = 0.0f;
    for (int w = 0; w < WAVES_PER_BLOCK; ++w) { S += wsum[w]; Cv += wcnt[w]; }
    out[0] = -S / Cv;
  }
}

// ===========================================================================
extern "C" void kernel_launch(void *const *d_in, const int *in_sizes, int n_in,
                              void *d_out, int out_size, void *d_ws,
                              size_t ws_size, hipStream_t stream) {
  const float *xn = (const float *)d_in[0];
  const float *xb = (const float *)d_in[1];
  const int *tgt = (const int *)d_in[2];
  const float *lam = (const float *)d_in[3];
  float *out = (float *)d_out;

  const long long total = (long long)in_sizes[0];  // N*C
  const int N = in_sizes[2];
  const long long C = total / (long long)N;
  const long long n4 = total / 4;

  float *ws = (float *)d_ws;
  float *part_m = ws;
  float *part_s = ws + NBLOCKS;
  float *lse = ws + 2 * NBLOCKS;

  celc_lse_partial<<<NBLOCKS, TPB, 0, stream>>>(xn, xb, lam, part_m, part_s,
                                                n4, total);
  celc_lse_final<<<1, TPB, 0, stream>>>(part_m, part_s, lse, NBLOCKS);
  celc_gather_loss<<<1, TPB, 0, stream>>>(xn, xb, tgt, lam, lse, out, N, C);

  (void)n_in; (void)out_size; (void)ws_size;
}